// Embedding_18957985644926
// MI455X (gfx1250) — compile-verified
//
#include <hip/hip_runtime.h>
#include <math.h>

// ---------------------------------------------------------------------------
// RGCN forward for MI455X (gfx1250, wave32).
// Dense math: V_WMMA_F32_16X16X4_F32 (full fp32, RNE accumulate)
// Data movement: A operand streamed HBM -> LDS with GLOBAL_LOAD_ASYNC_TO_LDS
// (ASYNCcnt, double-buffered, prefetch next chunk during compute); B operand
// (L2-resident weights) staged into a K-pair-interleaved LDS layout so every
// WMMA B fragment is a single conflict-free ds_load_b64.
// Aggregation: per-relation scatter-add via global_atomic_add_f32.
// ---------------------------------------------------------------------------

typedef __attribute__((ext_vector_type(2))) float v2f;
typedef __attribute__((ext_vector_type(8))) float v8f;

#define TILE_M 32
#define TILE_N 128
#define TILE_K 32
#define BPAD   (TILE_N + 16)   // pair-row stride 144*2 dwords == 32 mod 64 banks
#define GEMM_THREADS 128       // 4 wave32s

#define GF_BIAS 1
#define GF_TANH 2
#define GF_CIN  4

#if __has_builtin(__builtin_amdgcn_global_load_async_to_lds_b128)
#define HAVE_ASYNC_LDS 1
#define AS1 __attribute__((address_space(1)))
#define AS3 __attribute__((address_space(3)))
typedef int v4i __attribute__((vector_size(4 * sizeof(int))));
#define WAIT_ASYNCCNT(n) asm volatile("s_wait_asynccnt %0" ::"n"(n) : "memory")
#else
#define HAVE_ASYNC_LDS 0
#define WAIT_ASYNCCNT(n)
#endif

// Stage one 32x32 A chunk: 256 float4, 2 per thread, contiguous 16B both sides.
__device__ __forceinline__
void stage_A(const float* __restrict__ A, int rowBlk, int k0, int lda,
             float (*__restrict__ As)[TILE_K + 4], int tid)
{
#if HAVE_ASYNC_LDS
    #pragma unroll
    for (int it = 0; it < 2; ++it) {
        const int f  = tid + it * GEMM_THREADS;
        const int r  = f >> 3;
        const int c4 = f & 7;
        const float* gp = A + (size_t)(rowBlk + r) * lda + k0 + c4 * 4;
        __builtin_amdgcn_global_load_async_to_lds_b128(
            (AS1 v4i*)gp, (AS3 v4i*)&As[r][c4 * 4], 0, 0);
    }
#else
    float4 v[2];
    #pragma unroll
    for (int it = 0; it < 2; ++it) {
        const int f  = tid + it * GEMM_THREADS;
        const int r  = f >> 3;
        const int c4 = f & 7;
        v[it] = *(const float4*)(A + (size_t)(rowBlk + r) * lda + k0 + c4 * 4);
    }
    #pragma unroll
    for (int it = 0; it < 2; ++it) {
        const int f  = tid + it * GEMM_THREADS;
        *(float4*)(&As[f >> 3][(f & 7) * 4]) = v[it];
    }
#endif
}

// Stage one 32x128 B chunk into pair-interleaved layout:
// Bs[kp][n][v] = B[k0 + 2*kp + v][colBlk + n].  One column per thread
// (coalesced across lanes per row; B is tiny and L2-resident), then 16
// conflict-free ds_store_b64 (lane n -> dword 2n stride).
__device__ __forceinline__
void stage_B(const float* __restrict__ B, int colBlk, int k0, int ldb,
             float (*__restrict__ Bs)[BPAD][2], int n)
{
    float rb[TILE_K];
    #pragma unroll
    for (int k = 0; k < TILE_K; ++k)
        rb[k] = B[(size_t)(k0 + k) * ldb + colBlk + n];
    #pragma unroll
    for (int kp = 0; kp < TILE_K / 2; ++kp) {
        v2f w;
        w.x = rb[2 * kp + 0];
        w.y = rb[2 * kp + 1];
        *(v2f*)(&Bs[kp][n][0]) = w;
    }
}

// D[M,N] = A[M,K] @ B[K,N] (+ bias) (+ Cin) (tanh).
// Requires M % 32 == 0, N % 128 == 0, K % 32 == 0 (true for all shapes here).
__global__ __launch_bounds__(GEMM_THREADS)
void gemm_wmma_f32(const float* __restrict__ A, const float* __restrict__ B,
                   const float* __restrict__ Cin, const float* __restrict__ bias,
                   float* __restrict__ D,
                   int M, int N, int K, int lda, int ldb, int flags)
{
    __shared__ float As[2][TILE_M][TILE_K + 4];        // 9.2 KB
    __shared__ float Bs[2][TILE_K / 2][BPAD][2];       // 36.9 KB

    const int tid  = threadIdx.x;
    const int wave = tid >> 5;        // 0..3
    const int lane = tid & 31;
    const int lm   = lane & 15;       // M-row / N-col within a 16x16 tile
    const int lh   = lane >> 4;       // half-wave select

    const int rowBlk  = blockIdx.x * TILE_M;
    const int colBlk  = blockIdx.y * TILE_N;
    const int waveRow = (wave & 1) * 16;
    const int waveCol = (wave >> 1) * 64;

    v8f acc[4] = {};
    (void)M;

    const int kChunks = K / TILE_K;
    stage_A(A, rowBlk, 0, lda, As[0], tid);
    stage_B(B, colBlk, 0, ldb, Bs[0], tid);

    int buf = 0;
    for (int kc = 0; kc < kChunks; ++kc) {
        const int nxt = kc + 1;
        if (nxt < kChunks) {
            // prefetch next chunk into the other buffer while computing this one
            stage_A(A, rowBlk, nxt * TILE_K, lda, As[buf ^ 1], tid);
            stage_B(B, colBlk, nxt * TILE_K, ldb, Bs[buf ^ 1], tid);
            WAIT_ASYNCCNT(2);     // current buffer's 2 async b128 are complete
        } else {
            WAIT_ASYNCCNT(0);
        }
        __syncthreads();

        #pragma unroll
        for (int kk = 0; kk < TILE_K; kk += 4) {
            const int kb = kk + lh * 2;          // this half-wave's K-pair base
            const int p  = (kk >> 1) + lh;       // pair row in Bs
            // A 16x4 f32: lane lm = M row, VGPR v holds K = kb + v.
            v2f a = *(const v2f*)(&As[buf][waveRow + lm][kb]);
            #pragma unroll
            for (int t = 0; t < 4; ++t) {
                v2f b = *(const v2f*)(&Bs[buf][p][waveCol + t * 16 + lm][0]);
                acc[t] = __builtin_amdgcn_wmma_f32_16x16x4_f32(
                    false, a, false, b, (short)0, acc[t], false, false);
            }
        }
        __syncthreads();
        buf ^= 1;
    }

    // Epilogue. C/D layout: VGPR i -> M = i + lh*8, N = lm (per 16x16 tile).
    #pragma unroll
    for (int t = 0; t < 4; ++t) {
        const int col = colBlk + waveCol + t * 16 + lm;
        const float bv = (flags & GF_BIAS) ? bias[col] : 0.0f;
        #pragma unroll
        for (int i = 0; i < 8; ++i) {
            const int row = rowBlk + waveRow + i + lh * 8;
            const size_t idx = (size_t)row * N + col;
            float v = acc[t][i] + bv;
            if (flags & GF_CIN)  v += Cin[idx];
            if (flags & GF_TANH) v = tanhf(v);
            D[idx] = v;
        }
    }
}

// agg[dst[e]] += tmp[src[e]] for edges with etype[e] == rel.
// One wave per edge: lane carries one float4 (128 floats per message).
__global__ __launch_bounds__(256)
void scatter_rel(const float* __restrict__ tmp, float* __restrict__ agg,
                 const int* __restrict__ src, const int* __restrict__ dst,
                 const int* __restrict__ etype, int E, int rel)
{
    const int e = blockIdx.x * 8 + (threadIdx.x >> 5);
    if (e >= E) return;
    if (etype[e] != rel) return;          // uniform across the wave
    const int lane = threadIdx.x & 31;
    const float4 v = ((const float4*)(tmp + (size_t)src[e] * 128))[lane];
    float* d = agg + (size_t)dst[e] * 128 + lane * 4;
    atomicAdd(d + 0, v.x);
    atomicAdd(d + 1, v.y);
    atomicAdd(d + 2, v.z);
    atomicAdd(d + 3, v.w);
}

extern "C" void kernel_launch(void* const* d_in, const int* in_sizes, int n_in,
                              void* d_out, int out_size, void* d_ws, size_t ws_size,
                              hipStream_t stream)
{
    (void)n_in; (void)out_size; (void)ws_size;

    const float* feats  = (const float*)d_in[0];
    const int*   src    = (const int*)  d_in[1];
    const int*   dst    = (const int*)  d_in[2];
    const int*   etype  = (const int*)  d_in[3];
    const float* W_in   = (const float*)d_in[4];
    const float* b_in   = (const float*)d_in[5];
    const float* W_rel  = (const float*)d_in[6];   // [10,8,128,128]
    const float* W_loop = (const float*)d_in[7];   // [10,128,128]
    const float* b_rel  = (const float*)d_in[8];   // [10,128]
    const float* W_u1   = (const float*)d_in[9];   // [10,256,256]
    const float* b_u1   = (const float*)d_in[10];  // [10,256]
    const float* W_u2   = (const float*)d_in[11];  // [10,384,128]
    const float* b_u2   = (const float*)d_in[12];  // [10,128]

    const int F = 64, H = 128, R = 8, L = 10;
    const int M = in_sizes[0] / F;   // 100000 nodes (multiple of 32)
    const int E = in_sizes[1];       // 600000 edges

    // Workspace: x0,x1,agg,tmp [M,128] + mid [M,256]  (~307 MB total)
    float* ws  = (float*)d_ws;
    float* x0  = ws;
    float* x1  = x0  + (size_t)M * H;
    float* agg = x1  + (size_t)M * H;
    float* tmp = agg + (size_t)M * H;
    float* mid = tmp + (size_t)M * H;

    const dim3 blk(GEMM_THREADS);
    auto gemm = [&](const float* A, const float* B, const float* Cin,
                    const float* bias, float* Dm, int Nout, int K,
                    int lda, int ldb, int flags) {
        dim3 grd(M / TILE_M, Nout / TILE_N);
        gemm_wmma_f32<<<grd, blk, 0, stream>>>(A, B, Cin, bias, Dm,
                                               M, Nout, K, lda, ldb, flags);
    };

    // x = tanh(feats @ W_in + b_in)
    gemm(feats, W_in, nullptr, b_in, x0, H, F, F, H, GF_BIAS | GF_TANH);

    float* x  = x0;
    float* xn = x1;
    const int scatterBlocks = (E + 7) / 8;

    for (int l = 0; l < L; ++l) {
        // msg = x @ W_loop + b_rel  +  segment_sum of per-relation messages
        gemm(x, W_loop + (size_t)l * H * H, nullptr, b_rel + (size_t)l * H,
             agg, H, H, H, H, GF_BIAS);
        for (int r = 0; r < R; ++r) {
            gemm(x, W_rel + ((size_t)l * R + r) * H * H, nullptr, nullptr,
                 tmp, H, H, H, H, 0);
            scatter_rel<<<scatterBlocks, 256, 0, stream>>>(tmp, agg, src, dst,
                                                           etype, E, r);
        }

        // mid = tanh([x, msg] @ W_u1 + b_u1), split into top/bottom GEMMs
        const float* Wu1 = W_u1 + (size_t)l * (2 * H) * (2 * H);
        gemm(x,   Wu1,                       nullptr, b_u1 + (size_t)l * 2 * H,
             mid, 2 * H, H, H, 2 * H, GF_BIAS);
        gemm(agg, Wu1 + (size_t)H * (2 * H), mid,     nullptr,
             mid, 2 * H, H, H, 2 * H, GF_TANH | GF_CIN);

        // x' = tanh([x, mid] @ W_u2 + b_u2)
        const float* Wu2 = W_u2 + (size_t)l * (3 * H) * H;
        gemm(x,   Wu2,               nullptr, b_u2 + (size_t)l * H,
             tmp, H, H, H, H, GF_BIAS);
        float* dstx = (l == L - 1) ? (float*)d_out : xn;
        gemm(mid, Wu2 + (size_t)H * H, tmp,   nullptr,
             dstx, H, 2 * H, 2 * H, H, GF_TANH | GF_CIN);

        float* sw = x; x = xn; xn = sw;
    }
}